// Get_uncertain_pt_coords_randommness_17961553232034
// MI455X (gfx1250) — compile-verified
//
#include <hip/hip_runtime.h>
#include <stdint.h>

// Problem constants (match the JAX reference)
#define Bn 64
#define Hh 128
#define Ww 128
#define Cc 21
#define Pn 49152          // points sampled per batch
#define Kn 36864          // top-k kept
#define Rn 12288          // extra random appended
#define THREADS 1024      // 32 wave32 waves per workgroup
#define NPT (Pn / THREADS)  // 48 elements per thread
#define NDIG 16             // 4-bit digits, 8 passes
#define SMEM_WORDS (Pn + NDIG * THREADS + THREADS)   // keys + hist + scan tmp
#define SMEM_BYTES (SMEM_WORDS * 4)                  // 266,240 B <= 320KB/WG (CDNA5)

typedef unsigned int u32;

#if __has_builtin(__builtin_amdgcn_tensor_load_to_lds) && __has_builtin(__builtin_amdgcn_s_wait_tensorcnt)
#define USE_TDM 1
typedef unsigned int u32x4 __attribute__((ext_vector_type(4)));
typedef int          i32x4 __attribute__((ext_vector_type(4)));
typedef int          i32x8 __attribute__((ext_vector_type(8)));
#endif

// ---------------------------------------------------------------------------
// Kernel 1: bilinear sample channel 0, produce sortable key + point index.
// key = ~orderable(-|v|)  ==> ascending radix sort == descending uncertainty,
// stable sort == lower-index-first tie break (matches jax.lax.top_k).
// ---------------------------------------------------------------------------
__global__ __launch_bounds__(256) void sample_uncertainty_kernel(
    const float* __restrict__ logits,
    const float* __restrict__ rc,
    u32* __restrict__ keys,
    u32* __restrict__ vals)
{
    int gid = blockIdx.x * 256 + threadIdx.x;   // grid is exactly Bn*Pn threads
    int b = gid / Pn;
    int p = gid - b * Pn;

    float2 xy = ((const float2*)rc)[gid];
    float x = xy.x * 127.0f;                    // * (H-1)
    float y = xy.y * 127.0f;                    // * (W-1)
    float x0f = floorf(x), y0f = floorf(y);
    float x1f = ceilf(x),  y1f = ceilf(y);
    int x0 = (int)x0f, y0 = (int)y0f;
    int x1 = (int)x1f, y1 = (int)y1f;
    float mx = x - x0f, my = y - y0f;
    float omx = 1.0f - mx, omy = 1.0f - my;

    const float* Lb = logits + (size_t)b * (Hh * Ww * Cc);  // channel 0 only
    float p1 = Lb[((size_t)x0 * Ww + y0) * Cc];
    float p2 = Lb[((size_t)x1 * Ww + y0) * Cc];
    float p3 = Lb[((size_t)x0 * Ww + y1) * Cc];
    float p4 = Lb[((size_t)x1 * Ww + y1) * Cc];

    float p12 = p1 * omx + p2 * mx;
    float p34 = p3 * omx + p4 * mx;
    float v   = p12 * omy + p34 * my;

    float u = -fabsf(v);                        // uncertainty (<= -0.0 always)
    u32 s = __float_as_uint(u);
    s = (s & 0x80000000u) ? ~s : (s | 0x80000000u);  // monotonic ascending map
    keys[gid] = ~s;                             // invert: ascending sort => descending u
    vals[gid] = (u32)p;
}

// ---------------------------------------------------------------------------
// Kernel 2: one 4-bit stable LSD radix pass. One workgroup per batch.
// Keys for the whole batch (192 KB) staged into LDS via a single TDM
// tensor_load_to_lds; 16x1024 digit histogram + workgroup scan in LDS.
// ---------------------------------------------------------------------------
__global__ __launch_bounds__(THREADS) void radix_pass_kernel(
    const u32* __restrict__ keysIn, const u32* __restrict__ valsIn,
    u32* __restrict__ keysOut,      u32* __restrict__ valsOut,
    int shift)
{
    extern __shared__ u32 smem[];
    u32* skeys = smem;                    // [Pn]            keys for this batch
    u32* hist  = smem + Pn;               // [NDIG*THREADS]  digit-major histogram
    u32* stmp  = hist + NDIG * THREADS;   // [THREADS]       scan scratch

    const int tid  = threadIdx.x;
    const u32 base = (u32)blockIdx.x * Pn;

#ifdef USE_TDM
    if (tid < 32) {  // wave 0 issues the DMA; EXEC is ignored by TDM
        unsigned long long ga = (unsigned long long)(uintptr_t)(keysIn + base);
        u32 lds = (u32)(uintptr_t)skeys;  // low 32 bits of generic ptr = LDS offset
        // D# group0: count=1 valid | lds_addr | global_addr[56:0] | type=2
        u32x4 g0 = { 1u, lds, (u32)ga,
                     ((u32)((ga >> 32) & 0x1FFFFFFull)) | 0x80000000u };
        // D# group1: data_size=4B; tensor_dim0=tile_dim0=Pn (0xC000);
        // tensor_dim1=tile_dim1=1; tile_dim2=0; dim0_stride=Pn; dim1_stride=0
        i32x8 g1 = { (int)(2u << 16),          // data_size=2 -> 4 bytes
                     (int)(0xC000u << 16),     // tensor_dim0 lo16 in [31:16]
                     (int)(1u << 16),          // tensor_dim0 hi16=0 | tensor_dim1 lo16=1
                     (int)(0xC000u << 16),     // tensor_dim1 hi16=0 | tile_dim0=0xC000
                     1,                        // tile_dim1=1 | tile_dim2=0
                     (int)Pn,                  // tensor_dim0_stride lo32
                     0, 0 };
        i32x4 gz4 = { 0, 0, 0, 0 };
        i32x8 gz8 = { 0, 0, 0, 0, 0, 0, 0, 0 };
        // 6-arg form (clang-23 / therock-10.0 headers): groups 0..3 + extra
        // int32x8 group + cache policy immediate.
        __builtin_amdgcn_tensor_load_to_lds(g0, g1, gz4, gz4, gz8, 0);
        __builtin_amdgcn_s_wait_tensorcnt(0);
    }
#else
    for (int i = tid; i < Pn; i += THREADS) skeys[i] = keysIn[base + i];
#endif
    __syncthreads();

    // ---- Phase 1: per-thread digit counts over a contiguous 48-elem chunk ----
    u32 cnt[NDIG];
#pragma unroll
    for (int d = 0; d < NDIG; ++d) cnt[d] = 0;
    const int myBase = tid * NPT;
#pragma unroll 8
    for (int i = 0; i < NPT; ++i) {
        u32 k = skeys[myBase + i];
        cnt[(k >> shift) & (NDIG - 1)] += 1;
    }
#pragma unroll
    for (int d = 0; d < NDIG; ++d) hist[d * THREADS + tid] = cnt[d];
    __syncthreads();

    // ---- Phase 2: exclusive scan of 16384 counters in (digit-major,
    // thread-minor) order => stable global offsets. Thread t serially scans
    // contiguous entries [t*16, t*16+16), then workgroup scan of partial sums.
    u32 loc[NDIG];
    {
        const int gbase = tid * NDIG;
        u32 run = 0;
#pragma unroll
        for (int i = 0; i < NDIG; ++i) { loc[i] = run; run += hist[gbase + i]; }
        stmp[tid] = run;
    }
    __syncthreads();
    for (int off = 1; off < THREADS; off <<= 1) {   // Hillis-Steele inclusive
        u32 add = 0;
        if (tid >= off) add = stmp[tid - off];
        __syncthreads();
        stmp[tid] += add;
        __syncthreads();
    }
    {
        const int gbase = tid * NDIG;
        u32 tb = (tid > 0) ? stmp[tid - 1] : 0u;    // exclusive base
#pragma unroll
        for (int i = 0; i < NDIG; ++i) hist[gbase + i] = tb + loc[i];
    }
    __syncthreads();

    // ---- Phase 3: stable scatter. Each thread bumps only its own hist slots. ----
    for (int i = 0; i < NPT; ++i) {
        u32 k = skeys[myBase + i];
        u32 d = (k >> shift) & (NDIG - 1);
        u32 slot = d * THREADS + (u32)tid;
        u32 pos = hist[slot];
        hist[slot] = pos + 1;
        keysOut[base + pos] = k;
        valsOut[base + pos] = valsIn[base + myBase + i];
    }
}

// ---------------------------------------------------------------------------
// Kernel 3: out[b, k<K] = rc[b, sortedIdx[b,k]];  out[b, k>=K] = extra[b, k-K]
// ---------------------------------------------------------------------------
__global__ __launch_bounds__(256) void gather_concat_kernel(
    const u32* __restrict__ sortedIdx,
    const float* __restrict__ rc,
    const float* __restrict__ extra,
    float* __restrict__ out)
{
    int gid = blockIdx.x * 256 + threadIdx.x;
    int b = gid / Pn;
    int k = gid - b * Pn;
    float2 v;
    if (k < Kn) {
        u32 idx = sortedIdx[(size_t)b * Pn + k];
        v = ((const float2*)rc)[(size_t)b * Pn + idx];
    } else {
        v = ((const float2*)extra)[(size_t)b * Rn + (k - Kn)];
    }
    ((float2*)out)[gid] = v;
}

// ---------------------------------------------------------------------------
extern "C" void kernel_launch(void* const* d_in, const int* in_sizes, int n_in,
                              void* d_out, int out_size, void* d_ws, size_t ws_size,
                              hipStream_t stream) {
    (void)in_sizes; (void)n_in; (void)out_size; (void)ws_size;

    const float* logits = (const float*)d_in[0];   // (B,H,W,C) f32
    const float* rc     = (const float*)d_in[1];   // (B,P,2)   f32
    const float* extra  = (const float*)d_in[2];   // (B,R,2)   f32
    float* out          = (float*)d_out;           // (B,P,2)   f32

    const size_t N = (size_t)Bn * Pn;              // 3,145,728
    // Workspace: 4 * N * 4B = ~50 MB ping-pong key/val buffers.
    u32* keysA = (u32*)d_ws;
    u32* valsA = keysA + N;
    u32* keysB = valsA + N;
    u32* valsB = keysB + N;

    sample_uncertainty_kernel<<<dim3((unsigned)(N / 256)), dim3(256), 0, stream>>>(
        logits, rc, keysA, valsA);

    u32 *ki = keysA, *vi = valsA, *ko = keysB, *vo = valsB;
    for (int pass = 0; pass < 8; ++pass) {
        radix_pass_kernel<<<dim3(Bn), dim3(THREADS), SMEM_BYTES, stream>>>(
            ki, vi, ko, vo, pass * 4);
        u32* t;
        t = ki; ki = ko; ko = t;
        t = vi; vi = vo; vo = t;
    }
    // 8 passes (even) => sorted indices back in valsA (== vi)

    gather_concat_kernel<<<dim3((unsigned)(N / 256)), dim3(256), 0, stream>>>(
        vi, rc, extra, out);
}